// DWT_2D_tiny_66924180407265
// MI455X (gfx1250) — compile-verified
//
#include <hip/hip_runtime.h>

#define IMG   256   // N*C = 8*32
#define H_IN  512
#define W_IN  512
#define H_OUT 256
#define W_OUT 256

#define GLOBAL_AS __attribute__((address_space(1)))
#define LDS_AS    __attribute__((address_space(3)))

typedef float f4 __attribute__((ext_vector_type(4)));
typedef int   v4i __attribute__((__vector_size__(4 * sizeof(int))));

// Each thread computes 4 consecutive output columns of one output row.
// Input needed per thread: 8 floats from row 2p and 8 floats from row 2p+1
// (two b128 beats per row), staged through LDS via the gfx1250 async DMA path.
__global__ __launch_bounds__(256, 1) void haar_ll_kernel(const float* __restrict__ x,
                                                         float* __restrict__ y) {
    __shared__ float smem[256 * 16];   // 64 bytes per thread = 16 KB per block

    const unsigned tid = blockIdx.x * 256u + threadIdx.x;
    const unsigned q4  = tid & 63u;           // which group of 4 output columns
    const unsigned p   = (tid >> 6) & 255u;   // output row
    const unsigned img = tid >> 14;           // flattened (n, c)

    const size_t in_off = (size_t)img * ((size_t)H_IN * W_IN)
                        + (size_t)(2u * p) * W_IN
                        + (size_t)q4 * 8u;
    const float* g0 = x + in_off;        // row 2p,   8 floats (32B, 16B-aligned)
    const float* g1 = g0 + W_IN;         // row 2p+1, 8 floats

    float* lds = &smem[threadIdx.x * 16u];

    f4 a0, a1, b0, b1;

#if __has_builtin(__builtin_amdgcn_global_load_async_to_lds_b128)
    // gfx1250 async DMA: global -> LDS, tracked by ASYNCcnt.
    // Builtin signature (from hipcc diagnostics): param 1 is AS(1) int4*,
    // param 2 is the LDS destination, params 3/4 are imm offset / imm cpol.
    __builtin_amdgcn_global_load_async_to_lds_b128(
        (GLOBAL_AS v4i*)g0,       (LDS_AS v4i*)(lds + 0),  0, 0);
    __builtin_amdgcn_global_load_async_to_lds_b128(
        (GLOBAL_AS v4i*)(g0 + 4), (LDS_AS v4i*)(lds + 4),  0, 0);
    __builtin_amdgcn_global_load_async_to_lds_b128(
        (GLOBAL_AS v4i*)g1,       (LDS_AS v4i*)(lds + 8),  0, 0);
    __builtin_amdgcn_global_load_async_to_lds_b128(
        (GLOBAL_AS v4i*)(g1 + 4), (LDS_AS v4i*)(lds + 12), 0, 0);
  #if __has_builtin(__builtin_amdgcn_s_wait_asynccnt)
    __builtin_amdgcn_s_wait_asynccnt(0);
  #else
    asm volatile("s_wait_asynccnt 0x0" ::: "memory");
  #endif
    a0 = *(const f4*)(lds + 0);
    a1 = *(const f4*)(lds + 4);
    b0 = *(const f4*)(lds + 8);
    b1 = *(const f4*)(lds + 12);
#else
    // Inline-asm fallback: same ISA path. Per CDNA5 ISA the
    // instruction offset is added to BOTH the LDS and global addresses, so the
    // second 16B beat of each row reuses the same address VGPRs with offset:16.
    unsigned l0 = (unsigned)(size_t)(LDS_AS float*)lds;   // LDS byte address
    unsigned l1 = l0 + 32u;
    asm volatile("global_load_async_to_lds_b128 %0, %1, off"
                 :: "v"(l0), "v"(g0) : "memory");
    asm volatile("global_load_async_to_lds_b128 %0, %1, off offset:16"
                 :: "v"(l0), "v"(g0) : "memory");
    asm volatile("global_load_async_to_lds_b128 %0, %1, off"
                 :: "v"(l1), "v"(g1) : "memory");
    asm volatile("global_load_async_to_lds_b128 %0, %1, off offset:16"
                 :: "v"(l1), "v"(g1) : "memory");
    asm volatile("s_wait_asynccnt 0x0" ::: "memory");
    a0 = *(const volatile f4*)(lds + 0);
    a1 = *(const volatile f4*)(lds + 4);
    b0 = *(const volatile f4*)(lds + 8);
    b1 = *(const volatile f4*)(lds + 12);
#endif

    f4 o;
    o.x = 0.5f * ((a0.x + a0.y) + (b0.x + b0.y));
    o.y = 0.5f * ((a0.z + a0.w) + (b0.z + b0.w));
    o.z = 0.5f * ((a1.x + a1.y) + (b1.x + b1.y));
    o.w = 0.5f * ((a1.z + a1.w) + (b1.z + b1.w));

    const size_t out_off = (size_t)img * ((size_t)H_OUT * W_OUT)
                         + (size_t)p * W_OUT
                         + (size_t)q4 * 4u;
    *(f4*)(y + out_off) = o;
}

extern "C" void kernel_launch(void* const* d_in, const int* in_sizes, int n_in,
                              void* d_out, int out_size, void* d_ws, size_t ws_size,
                              hipStream_t stream) {
    (void)in_sizes; (void)n_in; (void)d_ws; (void)ws_size; (void)out_size;
    const float* x = (const float*)d_in[0];   // (8,32,512,512) fp32
    float* y = (float*)d_out;                 // (8,32,256,256) fp32

    // 256 images * 256 rows * 64 quad-column groups = 4,194,304 threads
    const unsigned total_threads = IMG * H_OUT * (W_OUT / 4);
    dim3 block(256);
    dim3 grid(total_threads / 256);           // 16384 blocks, exact cover
    haar_ll_kernel<<<grid, block, 0, stream>>>(x, y);
}